// ConstraintEnforcementLayer_63831803953580
// MI455X (gfx1250) — compile-verified
//
#include <hip/hip_runtime.h>
#include <stdint.h>

typedef __attribute__((ext_vector_type(2))) float    v2f;
typedef __attribute__((ext_vector_type(4))) float    v4f;
typedef __attribute__((ext_vector_type(8))) float    v8f;
typedef __attribute__((ext_vector_type(4))) uint32_t u32x4;
typedef __attribute__((ext_vector_type(4))) int32_t  i32x4;
typedef __attribute__((ext_vector_type(8))) int32_t  i32x8;

#define EPSF 1e-7f
#define TILE 16
#define LDS_STRIDE 260   // 256 + 4 pad (matches TDM pad_interval=7/pad_amount=3 -> 1040B pitch)

// Ac[mm] = dot(A[mm,:], c) ; r[mm] = b[mm] - Ac[mm]
__global__ __launch_bounds__(256) void cel_prep(const float* __restrict__ A,
                                                const float* __restrict__ b,
                                                const float* __restrict__ c,
                                                float* __restrict__ r,
                                                float* __restrict__ Ac,
                                                int m, int n) {
    int mm = blockIdx.x * blockDim.x + threadIdx.x;
    if (mm >= m) return;
    float acc = 0.0f;
    const float* row = A + (size_t)mm * n;
    for (int k = 0; k < n; ++k) acc = fmaf(row[k], c[k], acc);
    Ac[mm] = acc;
    r[mm]  = b[mm] - acc;
}

// One wave (32 threads) per 16-row batch tile.
// A_dot = y @ A^T via V_WMMA_F32_16X16X4_F32 (then -Ac in epilogue); y tile is
// staged into LDS by the Tensor Data Mover with native padding for bank-conflict-free reads.
__global__ __launch_bounds__(32) void cel_main(const float* __restrict__ y,
                                               const float* __restrict__ A,
                                               const float* __restrict__ c,
                                               const float* __restrict__ r,
                                               const float* __restrict__ Ac,
                                               float* __restrict__ z,
                                               int m, int n) {
    __shared__ float ytile[TILE * LDS_STRIDE];   // raw y tile, 1040B row pitch
    __shared__ float alpha_s[TILE];

    const int t = threadIdx.x;                   // 0..31 (one wave32)
    const int rowBase = blockIdx.x * TILE;

    // ---- TDM: async 2D tile load y[rowBase:rowBase+16, 0:n] -> LDS ----
    const uint64_t gaddr = (uint64_t)(uintptr_t)(y + (size_t)rowBase * n);
    const uint32_t laddr = (uint32_t)(uintptr_t)&ytile[0];   // flat LDS addr, low 32 = LDS offset

    u32x4 g0;
    g0[0] = 1u;                                           // count=1, is_restore=0, gather off
    g0[1] = laddr;                                        // lds_addr
    g0[2] = (uint32_t)(gaddr & 0xffffffffu);              // global_addr[31:0]
    g0[3] = (uint32_t)((gaddr >> 32) & 0x01ffffffu)       // global_addr[56:32]
          | (2u << 30);                                   // type=2 ("image")

    i32x8 g1;
    g1[0] = (int32_t)((2u << 16)        // data_size = 4B
                    | (1u << 20)        // pad_enable
                    | (7u << 22)        // pad_interval: every 256 DWORDs
                    | (3u << 25));      // pad_amount: 4 DWORDs -> 260-float pitch
    g1[1] = (int32_t)(((uint32_t)n & 0xffffu) << 16);     // tensor_dim0[15:0] @ bits 63:48
    g1[2] = (int32_t)((((uint32_t)n >> 16) & 0xffffu)     // tensor_dim0[31:16]
                    | ((uint32_t)TILE << 16));            // tensor_dim1[15:0] = 16 rows
    g1[3] = (int32_t)(((uint32_t)n & 0xffffu) << 16);     // tile_dim0 = n @ bits 127:112
    g1[4] = (int32_t)TILE;                                // tile_dim1 = 16
    g1[5] = (int32_t)n;                                   // tensor_dim0_stride[31:0]
    g1[6] = 0;                                            // stride hi / dim1_stride
    g1[7] = 0;

    i32x4 gz = {0, 0, 0, 0};
#if __has_include(<hip/amd_detail/amd_gfx1250_TDM.h>)
    i32x8 gz8 = {0, 0, 0, 0, 0, 0, 0, 0};                 // clang-23 / therock-10.0: 6-arg form
    __builtin_amdgcn_tensor_load_to_lds(g0, g1, gz, gz, gz8, 0);
#else
    __builtin_amdgcn_tensor_load_to_lds(g0, g1, gz, gz, 0);  // ROCm 7.2 clang-22: 5-arg form
#endif
    __builtin_amdgcn_s_wait_tensorcnt(0);                 // tile resident in LDS

    const int lane = t & 31;
    const int half = lane >> 4;               // 0: K={0,1}, 1: K={2,3}
    const int lrow = lane & 15;

    // A-operand (16x4 fp32): lane supplies row=lrow, K = 4k + 2*half + {0,1}
    const float* afrag = ytile + lrow * LDS_STRIDE + 2 * half;

    float runmin[8];
    #pragma unroll
    for (int g = 0; g < 8; ++g) runmin[g] = 2.0f;  // cand is always <= 2

    const int ksteps = n >> 2;                // 64 WMMA K-steps
    const int jtiles = m >> 4;                // 16 output m-tiles
    for (int j = 0; j < jtiles; ++j) {
        // B-operand (4x16 fp32 = A^T tile): lane = col N=lrow, K = 4k + 2*half + {0,1}
        const float* bfrag = A + (size_t)(j * TILE + lrow) * n + 2 * half;

        if (j + 1 < jtiles)   // uniform; warm L0/L2 for next A tile
            __builtin_prefetch(A + (size_t)((j + 1) * TILE + lrow) * n, 0, 1);

        v8f acc = {};
        #pragma unroll 8
        for (int k = 0; k < ksteps; ++k) {
            v2f a = *(const v2f*)(afrag + 4 * k);   // ds_load_b64
            v2f b = *(const v2f*)(bfrag + 4 * k);   // global_load_b64 (L2-resident)
            acc = __builtin_amdgcn_wmma_f32_16x16x4_f32(
                /*neg_a=*/false, a, /*neg_b=*/false, b,
                /*c_mod=*/(short)0, acc, /*reuse_a=*/false, /*reuse_b=*/false);
        }

        // A_dot = acc - Ac[m] ; ip = r[m]/(A_dot+eps) ; cand filter ; running min
        float rv = r[j * TILE + lrow];
        float av = Ac[j * TILE + lrow];
        #pragma unroll
        for (int g = 0; g < 8; ++g) {
            float ip   = rv / ((acc[g] - av) + EPSF);
            float cand = (ip > 1.0f || ip < 0.0f) ? 2.0f : ip;
            runmin[g]  = fminf(runmin[g], cand);
        }
    }

    // min over the 16 N-lanes inside each half (xor masks 1..8 stay in-half)
    #pragma unroll
    for (int g = 0; g < 8; ++g) {
        #pragma unroll
        for (int mask = 1; mask < 16; mask <<= 1)
            runmin[g] = fminf(runmin[g], __shfl_xor(runmin[g], mask, 32));
    }

    // C/D layout: VGPR g, half h -> batch row g + 8*h
    if (lrow == 0) {
        #pragma unroll
        for (int g = 0; g < 8; ++g) {
            float a = runmin[g];
            alpha_s[half * 8 + g] = (a > 1.0f) ? 1.0f : a;
        }
    }
    __syncthreads();

    // z = alpha*y + (1-alpha)*c, y re-read from the LDS tile (one HBM read of y total)
    const int nv4 = n >> 2;
    const int total4 = TILE * nv4;
    for (int f = t; f < total4; f += 32) {
        int row = f / nv4;
        int col = (f % nv4) << 2;
        float al = alpha_s[row];
        v4f yv = *(const v4f*)(ytile + row * LDS_STRIDE + col);
        v4f cv = *(const v4f*)(c + col);
        v4f zv = al * yv + (1.0f - al) * cv;
        *(v4f*)(z + (size_t)(rowBase + row) * n + col) = zv;
    }
}

extern "C" void kernel_launch(void* const* d_in, const int* in_sizes, int n_in,
                              void* d_out, int out_size, void* d_ws, size_t ws_size,
                              hipStream_t stream) {
    const float* y = (const float*)d_in[0];   // [B, n]
    const float* A = (const float*)d_in[1];   // [m, n]
    const float* b = (const float*)d_in[2];   // [m]
    const float* c = (const float*)d_in[3];   // [n]
    float* z = (float*)d_out;                 // [B, n]

    const int m = in_sizes[2];
    const int n = in_sizes[3];
    const int B = in_sizes[0] / n;

    float* r  = (float*)d_ws;                 // [m]
    float* Ac = (float*)d_ws + m;             // [m]

    cel_prep<<<(m + 255) / 256, 256, 0, stream>>>(A, b, c, r, Ac, m, n);
    cel_main<<<B / TILE, 32, 0, stream>>>(y, A, c, r, Ac, z, m, n);
}